// GCNDetector_24455543783495
// MI455X (gfx1250) — compile-verified
//
#include <hip/hip_runtime.h>
#include <hip/hip_bf16.h>

typedef __attribute__((ext_vector_type(2))) float v2f;
typedef __attribute__((ext_vector_type(8))) float v8f;

#define KDIM 256          // inner dim is 256 for all three layers
#define KP   (KDIM + 4)   // A-tile LDS row pad: rows -> banks 4 apart, conflict-free
#define KC   32           // K chunk staged per iteration
#define WP   264          // W-tile LDS row pad: 2*WP % 64 == 16 -> half-waves use disjoint bank spans

// ---------------- degree / normalization ----------------
__global__ void deg_init(float* deg, int n) {
    int i = blockIdx.x * blockDim.x + threadIdx.x;
    if (i < n) deg[i] = 1.0f;                 // self-loop
}
__global__ void deg_count(const int* __restrict__ dst, float* deg, int e) {
    int i = blockIdx.x * blockDim.x + threadIdx.x;
    if (i < e) atomicAdd(&deg[dst[i]], 1.0f);
}
__global__ void deg_rsqrt(float* deg, int n) {
    int i = blockIdx.x * blockDim.x + threadIdx.x;
    if (i < n) deg[i] = rsqrtf(deg[i]);
}

// ---------------- GEMM: C[M x Nout] = (relu?)A[M x 256] * W[256 x Nout] ----------------
// grid = M/16 blocks, 256 threads = 8 waves. Block computes 16 rows x Nout cols.
// Wave w owns cols {w*16 .. w*16+15} (+128 for the second tile when NT==2).
// A tile (16x256) staged once in LDS; W staged in 32-row k-chunks (coalesced float4),
// so every WMMA operand comes from LDS.
template <int NT>
__global__ __launch_bounds__(256)
void gemm_wmma(const float* __restrict__ A, const float* __restrict__ W,
               float* __restrict__ C, int noutShift, int applyRelu) {
    __shared__ float As[16 * KP];   // 16.6 KB
    __shared__ float Ws[KC * WP];   // 33.8 KB
    const int Nout = 1 << noutShift;
    const int rowBase = blockIdx.x * 16;
    const int tid = threadIdx.x;

    // Stage contiguous 16x256 A tile (fuse ReLU of previous layer).
    {
        const float4* Ag = (const float4*)(A + (size_t)rowBase * KDIM);
        for (int i4 = tid; i4 < (16 * KDIM) / 4; i4 += 256) {
            float4 v = Ag[i4];
            if (applyRelu) {
                v.x = fmaxf(v.x, 0.0f); v.y = fmaxf(v.y, 0.0f);
                v.z = fmaxf(v.z, 0.0f); v.w = fmaxf(v.w, 0.0f);
            }
            int idx = i4 << 2;
            *(float4*)&As[(idx >> 8) * KP + (idx & 255)] = v;
        }
    }

    const int wave  = tid >> 5;
    const int lane  = tid & 31;
    const int lhalf = lane >> 4;       // 0: K parity {0,1}; 1: K parity {2,3}
    const int l16   = lane & 15;

    const float* Ap = As + l16 * KP + 2 * lhalf;   // A[M=l16][k + 2*lhalf]
    v8f acc[NT] = {};

    for (int k0 = 0; k0 < KDIM; k0 += KC) {
        __syncthreads();   // previous chunk fully consumed (also makes As visible)
        // Stage W rows k0..k0+31 (contiguous 32*Nout floats), coalesced float4.
        {
            const float4* Wg = (const float4*)(W + ((size_t)k0 << noutShift));
            const int n4 = (KC << noutShift) >> 2;
            const int cmask = Nout - 1;
            for (int i4 = tid; i4 < n4; i4 += 256) {
                int idx = i4 << 2;
                *(float4*)&Ws[(idx >> noutShift) * WP + (idx & cmask)] = Wg[i4];
            }
        }
        __syncthreads();

        #pragma unroll
        for (int kk = 0; kk < KC; kk += 4) {
            v2f a = { Ap[k0 + kk], Ap[k0 + kk + 1] };        // K = k+2h, k+2h+1
            const float* Wrow = Ws + (kk + 2 * lhalf) * WP + wave * 16 + l16;
            #pragma unroll
            for (int t = 0; t < NT; ++t) {
                v2f b = { Wrow[t * 128], Wrow[WP + t * 128] };
                acc[t] = __builtin_amdgcn_wmma_f32_16x16x4_f32(
                             false, a, false, b, (short)0, acc[t], false, false);
            }
        }
    }

    // C/D layout: VGPR r -> row r (lanes 0-15) / row 8+r (lanes 16-31)
    #pragma unroll
    for (int t = 0; t < NT; ++t) {
        int col = wave * 16 + t * 128 + l16;
        #pragma unroll
        for (int r = 0; r < 8; ++r) {
            int row = rowBase + r + 8 * lhalf;
            C[((size_t)row << noutShift) + col] = acc[t][r];
        }
    }
}

// ---------------- aggregation ----------------
// out[i][f] = h[i][f]*dinv[i]^2 + bias[f]   (writes every element -> safe vs poison)
__global__ void agg_init(const float* __restrict__ h, const float* __restrict__ bias,
                         const float* __restrict__ dinv, float* __restrict__ out,
                         long total, int fshift, int fmask) {
    long i = blockIdx.x * (long)blockDim.x + threadIdx.x;
    if (i >= total) return;
    int node = (int)(i >> fshift);
    int f = (int)i & fmask;
    float di = dinv[node];
    out[i] = h[i] * di * di + bias[f];
}

// out[dst][f] += h[src][f] * dinv[src]*dinv[dst]; F/4 threads per edge, float4 reads.
__global__ void agg_scatter(const float* __restrict__ h, const int* __restrict__ src,
                            const int* __restrict__ dst, const float* __restrict__ dinv,
                            float* __restrict__ out, long total, int gshift, int gmask,
                            int fdim) {
    long t = blockIdx.x * (long)blockDim.x + threadIdx.x;
    if (t >= total) return;
    long e = t >> gshift;
    int  g = (int)t & gmask;
    int  s = src[e], d = dst[e];
    float nrm = dinv[s] * dinv[d];
    const float4* hp = (const float4*)(h + (size_t)s * fdim);
    float4 v = hp[g];
    float* op = out + (size_t)d * fdim + g * 4;
    atomicAdd(op + 0, v.x * nrm);
    atomicAdd(op + 1, v.y * nrm);
    atomicAdd(op + 2, v.z * nrm);
    atomicAdd(op + 3, v.w * nrm);
}

extern "C" void kernel_launch(void* const* d_in, const int* in_sizes, int n_in,
                              void* d_out, int out_size, void* d_ws, size_t ws_size,
                              hipStream_t stream) {
    const float* x  = (const float*)d_in[0];
    const int*   ei = (const int*)d_in[1];
    const float* W1 = (const float*)d_in[2]; const float* b1 = (const float*)d_in[3];
    const float* W2 = (const float*)d_in[4]; const float* b2 = (const float*)d_in[5];
    const float* W3 = (const float*)d_in[6]; const float* b3 = (const float*)d_in[7];

    const int INF = 256, H = 256, OUTF = 128;
    const int  N = in_sizes[0] / INF;     // 50000 (divisible by 16)
    const long E = in_sizes[1] / 2;
    const int* srcI = ei;
    const int* dstI = ei + E;

    float* ws   = (float*)d_ws;
    float* dinv = ws;                                       // N floats
    float* bufA = ws + (((size_t)N + 255) & ~(size_t)255);  // N*256
    float* bufB = bufA + (size_t)N * H;                     // N*256
    float* outp = (float*)d_out;                            // N*128

    const int T = 256;
    // degrees -> dinv
    deg_init <<<(N + T - 1) / T, T, 0, stream>>>(dinv, N);
    deg_count<<<(int)((E + T - 1) / T), T, 0, stream>>>(dstI, dinv, (int)E);
    deg_rsqrt<<<(N + T - 1) / T, T, 0, stream>>>(dinv, N);

    long totN256 = (long)N * H;
    long totN128 = (long)N * OUTF;
    long totE256 = E << 6;   // E * 64 groups of 4 floats
    long totE128 = E << 5;   // E * 32 groups

    // ---- layer 1: h1 = x @ W1 ; B = agg(h1) + b1 ----
    gemm_wmma<2><<<N / 16, 256, 0, stream>>>(x, W1, bufA, 8, 0);
    agg_init <<<(int)((totN256 + T - 1) / T), T, 0, stream>>>(bufA, b1, dinv, bufB, totN256, 8, 255);
    agg_scatter<<<(int)((totE256 + T - 1) / T), T, 0, stream>>>(bufA, srcI, dstI, dinv, bufB, totE256, 6, 63, H);

    // ---- layer 2: h2 = relu(B) @ W2 ; B = agg(h2) + b2 ----
    gemm_wmma<2><<<N / 16, 256, 0, stream>>>(bufB, W2, bufA, 8, 1);
    agg_init <<<(int)((totN256 + T - 1) / T), T, 0, stream>>>(bufA, b2, dinv, bufB, totN256, 8, 255);
    agg_scatter<<<(int)((totE256 + T - 1) / T), T, 0, stream>>>(bufA, srcI, dstI, dinv, bufB, totE256, 6, 63, H);

    // ---- layer 3: h3 = relu(B) @ W3 ; out = agg(h3) + b3 ----
    gemm_wmma<1><<<N / 16, 256, 0, stream>>>(bufB, W3, bufA, 7, 1);
    agg_init <<<(int)((totN128 + T - 1) / T), T, 0, stream>>>(bufA, b3, dinv, outp, totN128, 7, 127);
    agg_scatter<<<(int)((totE128 + T - 1) / T), T, 0, stream>>>(bufA, srcI, dstI, dinv, outp, totE128, 5, 31, OUTF);
}